// LinearQuadPool2d_3762391351408
// MI455X (gfx1250) — compile-verified
//
#include <hip/hip_runtime.h>

namespace {

typedef float v4f __attribute__((ext_vector_type(4)));
typedef float v2f __attribute__((ext_vector_type(2)));

constexpr float kScale = 1.0e7f;      // 10**PRECISION
constexpr float kX0 = -180.0f;
constexpr float kY0 = -90.0f;
constexpr float kW  = 360.0f;
constexpr float kH  = 180.0f;
constexpr int   kG  = 512;            // 1 << MAX_DEPTH

// Spread 9 low bits of v to even bit positions (Morton part1by1).
__device__ __forceinline__ unsigned spread9(unsigned v) {
    v &= 0x1FFu;
    v = (v | (v << 8)) & 0x00FF00FFu;
    v = (v | (v << 4)) & 0x0F0F0F0Fu;
    v = (v | (v << 2)) & 0x33333333u;
    v = (v | (v << 1)) & 0x55555555u;
    return v;
}

__device__ __forceinline__ int quad_index(float cx, float cy) {
    // Exact replication of the reference fp32 chain:
    // round to 7 decimals (RNE), normalize by divide, quantize, clamp.
    float rx = rintf(cx * kScale) / kScale;
    float ry = rintf(cy * kScale) / kScale;
    float u = (rx - kX0) / kW;
    float v = (ry - kY0) / kH;
    int ix = (int)(u * (float)kG);   // truncation toward zero == astype(int32)
    int iy = (int)(v * (float)kG);
    ix = min(max(ix, 0), kG - 1);
    iy = min(max(iy, 0), kG - 1);
    return (int)(spread9((unsigned)ix) | (spread9((unsigned)iy) << 1));
}

// Prologue: interleave weight/bias into float2 table (2 MB, L2-resident).
__global__ __launch_bounds__(256) void pack_wb_kernel(const float* __restrict__ w,
                                                      const float* __restrict__ b,
                                                      v2f* __restrict__ wb, int n) {
    int i = blockIdx.x * blockDim.x + threadIdx.x;
    if (i < n) {
        v2f p;
        p.x = w[i];
        p.y = b[i];
        wb[i] = p;
    }
}

// Main kernel, packed-table path: 4 elements per thread, NT b128 stream I/O,
// one b64 gather per element (RT policy -> table stays hot in L2).
__global__ __launch_bounds__(256, 8) void quad_affine_packed(
    const v4f* __restrict__ coords,   // 2 points per v4f
    const v4f* __restrict__ xv,
    const v2f* __restrict__ wb,
    v4f* __restrict__ out, int n4, int n)
{
    int t = blockIdx.x * blockDim.x + threadIdx.x;
    if (t >= n4) return;

    // Stream prefetch ahead of the wavefront (gfx1250 global_prefetch_b8;
    // speculative — OOB translation failures are silently dropped).
    __builtin_prefetch((const char*)(coords + 2 * t) + 16384, 0, 0);
    __builtin_prefetch((const char*)(xv + t) + 8192, 0, 0);

    int e0 = t << 2;
    if (e0 + 3 < n) {
        // Read-once streams: non-temporal loads keep L2 free for the table.
        v4f c01 = __builtin_nontemporal_load(coords + 2 * t);
        v4f c23 = __builtin_nontemporal_load(coords + 2 * t + 1);
        v4f x4  = __builtin_nontemporal_load(xv + t);
        int i0 = quad_index(c01.x, c01.y);
        int i1 = quad_index(c01.z, c01.w);
        int i2 = quad_index(c23.x, c23.y);
        int i3 = quad_index(c23.z, c23.w);
        v2f p0 = wb[i0];
        v2f p1 = wb[i1];
        v2f p2 = wb[i2];
        v2f p3 = wb[i3];
        v4f o;
        o.x = fmaf(p0.x, x4.x, p0.y);
        o.y = fmaf(p1.x, x4.y, p1.y);
        o.z = fmaf(p2.x, x4.z, p2.y);
        o.w = fmaf(p3.x, x4.w, p3.y);
        __builtin_nontemporal_store(o, out + t);   // write-once stream
    } else {
        const float* cf = (const float*)coords;
        const float* xf = (const float*)xv;
        float*       of = (float*)out;
        for (int e = e0; e < n; ++e) {
            int idx = quad_index(cf[2 * e], cf[2 * e + 1]);
            v2f p = wb[idx];
            of[e] = fmaf(p.x, xf[e], p.y);
        }
    }
}

// Fallback path (no scratch): gather from the two separate tables.
__global__ __launch_bounds__(256, 8) void quad_affine_split(
    const v4f* __restrict__ coords,
    const v4f* __restrict__ xv,
    const float* __restrict__ w,
    const float* __restrict__ b,
    v4f* __restrict__ out, int n4, int n)
{
    int t = blockIdx.x * blockDim.x + threadIdx.x;
    if (t >= n4) return;
    __builtin_prefetch((const char*)(coords + 2 * t) + 16384, 0, 0);
    __builtin_prefetch((const char*)(xv + t) + 8192, 0, 0);

    int e0 = t << 2;
    if (e0 + 3 < n) {
        v4f c01 = __builtin_nontemporal_load(coords + 2 * t);
        v4f c23 = __builtin_nontemporal_load(coords + 2 * t + 1);
        v4f x4  = __builtin_nontemporal_load(xv + t);
        int i0 = quad_index(c01.x, c01.y);
        int i1 = quad_index(c01.z, c01.w);
        int i2 = quad_index(c23.x, c23.y);
        int i3 = quad_index(c23.z, c23.w);
        v4f o;
        o.x = fmaf(w[i0], x4.x, b[i0]);
        o.y = fmaf(w[i1], x4.y, b[i1]);
        o.z = fmaf(w[i2], x4.z, b[i2]);
        o.w = fmaf(w[i3], x4.w, b[i3]);
        __builtin_nontemporal_store(o, out + t);
    } else {
        const float* cf = (const float*)coords;
        const float* xf = (const float*)xv;
        float*       of = (float*)out;
        for (int e = e0; e < n; ++e) {
            int idx = quad_index(cf[2 * e], cf[2 * e + 1]);
            of[e] = fmaf(w[idx], xf[e], b[idx]);
        }
    }
}

} // namespace

extern "C" void kernel_launch(void* const* d_in, const int* in_sizes, int n_in,
                              void* d_out, int out_size, void* d_ws, size_t ws_size,
                              hipStream_t stream) {
    const v4f*   coords = (const v4f*)d_in[0];   // [N,2] f32 -> N/2 v4f
    const v4f*   xv     = (const v4f*)d_in[1];   // [N]   f32 -> N/4 v4f
    const float* w      = (const float*)d_in[2]; // [262144] f32
    const float* b      = (const float*)d_in[3]; // [262144] f32
    v4f*         out    = (v4f*)d_out;

    const int n  = in_sizes[1];        // N elements
    const int nt = in_sizes[2];        // table entries (4**MAX_DEPTH)
    const int n4 = (n + 3) >> 2;       // threads (4 elements each)

    const int threads = 256;
    const int blocks  = (n4 + threads - 1) / threads;

    if (ws_size >= (size_t)nt * sizeof(v2f)) {
        v2f* wb = (v2f*)d_ws;
        pack_wb_kernel<<<(nt + threads - 1) / threads, threads, 0, stream>>>(w, b, wb, nt);
        quad_affine_packed<<<blocks, threads, 0, stream>>>(coords, xv, wb, out, n4, n);
    } else {
        quad_affine_split<<<blocks, threads, 0, stream>>>(coords, xv, w, b, out, n4, n);
    }
}